// TCLoss_3994319586031
// MI455X (gfx1250) — compile-verified
//
#include <hip/hip_runtime.h>
#include <stdint.h>

// -----------------------------------------------------------------------------
// TCLoss (temporal-consistency mIoU loss), MI455X / gfx1250.
// Memory-bound streaming reduction (~343 MB @ 23.3 TB/s ~= 15 us floor).
// CDNA5 paths used: Tensor Data Mover (tensor_load_to_lds) for double-buffered
// global->LDS staging of `targets` tiles, s_wait_tensorcnt, global_prefetch_b8,
// wave32 shuffle reductions. No WMMA: workload has no GEMM structure.
// -----------------------------------------------------------------------------

#define IGNORE_INDEX 255

constexpr int B_ = 4, C_ = 19, H_ = 512, W_ = 1024;
constexpr int HW_ = H_ * W_;
constexpr int CHUNK = 256;               // pixels per chunk (== blockDim.x)
constexpr int NCHUNKS = HW_ / CHUNK;     // 2048
constexpr int GRID_X = 256;              // blocks per batch image
constexpr int CPB = NCHUNKS / GRID_X;    // 8 chunks per block
static_assert(NCHUNKS % GRID_X == 0, "chunking must divide evenly");

typedef __attribute__((ext_vector_type(4))) unsigned int u32x4;
typedef __attribute__((ext_vector_type(8))) int i32x8;
typedef __attribute__((ext_vector_type(4))) int i32x4;

// Issue one TDM load: 2D tile = C_ rows (channel planes, stride HW_ elems) x
// CHUNK contiguous f32 elements, packed contiguously into LDS at lds_byte.
// D# layout per CDNA5 ISA sec 8.3-8.6.
__device__ __forceinline__ void tdm_load_targets(const float* gsrc, uint32_t lds_byte) {
  const uint64_t ga = (uint64_t)(uintptr_t)gsrc;
  const u32x4 g0 = {
      1u,                                                // count=1, user D#, no gather
      lds_byte,                                          // lds_addr (bytes)
      (uint32_t)ga,                                      // global_addr[31:0]
      ((uint32_t)(ga >> 32) & 0x01FFFFFFu) | (2u << 30)  // global_addr[56:32] | type=2
  };
  // group1: data_size=2 (4B); tensor_dim0 huge (tiles always in-bounds),
  // tensor_dim1=C_; tile_dim0=CHUNK, tile_dim1=C_, tile_dim2=0;
  // tensor_dim0_stride=HW_ elements; dim1 stride unused for 2D tile.
  const i32x8 g1 = {
      (int)0x00020000u,                                  // data_size=2, mask=0, no flags
      0,                                                 // abar_addr=0 | td0[15:0]<<16 (=0)
      (int)((0x40000000u >> 16) | ((unsigned)C_ << 16)), // td0[31:16] | td1[15:0]<<16
      (int)((unsigned)CHUNK << 16),                      // td1[31:16](=0) | tile_dim0<<16
      (int)C_,                                           // tile_dim1 | tile_dim2<<16
      (int)HW_,                                          // tensor_dim0_stride[31:0]
      0, 0                                               // stride0[47:32] | stride1
  };
  const i32x4 g2 = {0, 0, 0, 0};
  const i32x4 g3 = {0, 0, 0, 0};
#if defined(__clang_major__) && (__clang_major__ >= 23)
  const i32x8 g4 = {0, 0, 0, 0, 0, 0, 0, 0};
  __builtin_amdgcn_tensor_load_to_lds(g0, g1, g2, g3, g4, 0);
#else
  __builtin_amdgcn_tensor_load_to_lds(g0, g1, g2, g3, 0);
#endif
}

__global__ __launch_bounds__(256) void tcloss_main(
    const float* __restrict__ preds,    // (B,C,H,W)
    const float* __restrict__ targets,  // (B,C,H,W)
    const float* __restrict__ flow,     // (B,1,H,W,2)
    const int*   __restrict__ labels,   // (B,H,W)
    float* __restrict__ pN,             // (B*C, GRID_X) partial num
    float* __restrict__ pD)             // (B*C, GRID_X) partial den
{
  __shared__ float buf[2][C_ * CHUNK];  // TDM double buffer: 2 x 19 KB
  __shared__ float sN[8][C_], sD[8][C_];

  const int tid  = threadIdx.x;
  const int lane = tid & 31;
  const int wv   = tid >> 5;
  const bool wave0 = (wv == 0);
  const int b = blockIdx.y;
  const size_t bHW  = (size_t)b * HW_;
  const size_t bCHW = (size_t)b * C_ * HW_;

  float num[C_], den[C_];
#pragma unroll
  for (int c = 0; c < C_; ++c) { num[c] = 0.0f; den[c] = 0.0f; }

  // Prologue: stage chunk 0 into buf[0].
  if (wave0) {
    tdm_load_targets(targets + bCHW + (size_t)blockIdx.x * CHUNK,
                     (uint32_t)(uintptr_t)&buf[0][0]);
    __builtin_amdgcn_s_wait_tensorcnt(0);
  }
  __syncthreads();

  for (int j = 0; j < CPB; ++j) {
    const int cid  = blockIdx.x + j * GRID_X;
    const int base = cid * CHUNK;
    const float* bufj = buf[j & 1];

    // Kick off async TDM for the next chunk into the other buffer.
    if (j + 1 < CPB) {
      if (wave0) {
        const int base2 = (blockIdx.x + (j + 1) * GRID_X) * CHUNK;
        tdm_load_targets(targets + bCHW + (size_t)base2,
                         (uint32_t)(uintptr_t)&buf[(j + 1) & 1][0]);
      }
      // Warm L2 for next chunk's preds gather (flow ~ N(0,1) => src ~= dst).
      const int p2 = (blockIdx.x + (j + 1) * GRID_X) * CHUNK + tid;
      const float* pp2 = preds + bCHW + (size_t)p2;
#pragma unroll
      for (int c = 0; c < C_; ++c)
        __builtin_prefetch(pp2 + (size_t)c * HW_, 0, 1);
    }

    // ---- compute current chunk ----
    const int p = base + tid;
    const float2 f = reinterpret_cast<const float2*>(flow)[bHW + p];  // (dx, dy)
    const int y = p >> 10;           // W_ = 1024
    const int x = p & (W_ - 1);
    const int ry = (int)rintf((float)y + f.y);   // round-half-even == jnp.round
    const int rx = (int)rintf((float)x + f.x);
    const bool valid = (ry >= 0) && (ry < H_) && (rx >= 0) && (rx < W_);
    const int ryc = min(max(ry, 0), H_ - 1);
    const int rxc = min(max(rx, 0), W_ - 1);

    // softmax(preds) gathered at the warped source pixel
    const float* ps = preds + bCHW + (size_t)ryc * W_ + rxc;
    float ep[C_];
    float mp = -3.402823466e38f;
#pragma unroll
    for (int c = 0; c < C_; ++c) { ep[c] = ps[(size_t)c * HW_]; mp = fmaxf(mp, ep[c]); }
    float sp = 0.0f;
#pragma unroll
    for (int c = 0; c < C_; ++c) { ep[c] = __expf(ep[c] - mp); sp += ep[c]; }
    const float pscale = valid ? (1.0f / sp) : 0.0f;   // warp fill value 0

    // softmax(targets) from the LDS-staged tile
    float et[C_];
    float mt = -3.402823466e38f;
#pragma unroll
    for (int c = 0; c < C_; ++c) { et[c] = bufj[c * CHUNK + tid]; mt = fmaxf(mt, et[c]); }
    float st = 0.0f;
#pragma unroll
    for (int c = 0; c < C_; ++c) { et[c] = __expf(et[c] - mt); st += et[c]; }
    const int lab = labels[bHW + p];
    const float tscale = (lab != IGNORE_INDEX) ? (1.0f / st) : 0.0f;

#pragma unroll
    for (int c = 0; c < C_; ++c) {
      const float pw = ep[c] * pscale;
      const float tt = et[c] * tscale;
      const float pt = pw * tt;
      num[c] += pt;                 // |pw*t|     (both >= 0)
      den[c] += (pw + tt) - pt;     // |pw+t-pw*t| (>= 0)
    }

    // Next chunk's TDM must have landed before anyone reads it; the barrier
    // also fences buffer reuse against the following iteration's TDM issue.
    if (wave0) __builtin_amdgcn_s_wait_tensorcnt(0);
    __syncthreads();
  }

  // ---- deterministic block reduction: wave32 shuffles -> LDS -> partials ----
#pragma unroll
  for (int c = 0; c < C_; ++c) {
    float n = num[c], d = den[c];
#pragma unroll
    for (int off = 16; off > 0; off >>= 1) {
      n += __shfl_xor(n, off);
      d += __shfl_xor(d, off);
    }
    if (lane == 0) { sN[wv][c] = n; sD[wv][c] = d; }
  }
  __syncthreads();
  if (tid < C_) {
    float n = 0.0f, d = 0.0f;
#pragma unroll
    for (int w = 0; w < 8; ++w) { n += sN[w][tid]; d += sD[w][tid]; }
    pN[(b * C_ + tid) * GRID_X + blockIdx.x] = n;
    pD[(b * C_ + tid) * GRID_X + blockIdx.x] = d;
  }
}

// One block per (b,c): reduce GRID_X(=256) partials, emit num/den ratio.
__global__ __launch_bounds__(256) void tcloss_reduce(
    const float* __restrict__ pN, const float* __restrict__ pD,
    float* __restrict__ r)
{
  const int bc = blockIdx.x;
  const int tid = threadIdx.x;
  float n = pN[bc * GRID_X + tid];
  float d = pD[bc * GRID_X + tid];
#pragma unroll
  for (int off = 16; off > 0; off >>= 1) {
    n += __shfl_xor(n, off);
    d += __shfl_xor(d, off);
  }
  __shared__ float wn[8], wd[8];
  if ((tid & 31) == 0) { wn[tid >> 5] = n; wd[tid >> 5] = d; }
  __syncthreads();
  if (tid == 0) {
    float N = 0.0f, D = 0.0f;
#pragma unroll
    for (int w = 0; w < 8; ++w) { N += wn[w]; D += wd[w]; }
    r[bc] = N / D;
  }
}

__global__ void tcloss_finalize(const float* __restrict__ r, float* __restrict__ out) {
  if (threadIdx.x == 0 && blockIdx.x == 0) {
    float acc = 0.0f;
    for (int b = 0; b < B_; ++b) {
      float m = 0.0f;
      for (int c = 0; c < C_; ++c) m += r[b * C_ + c];
      acc += 1.0f - m / (float)C_;
    }
    out[0] = acc / (float)B_;   // LOSS_WEIGHT = 1.0
  }
}

extern "C" void kernel_launch(void* const* d_in, const int* in_sizes, int n_in,
                              void* d_out, int out_size, void* d_ws, size_t ws_size,
                              hipStream_t stream) {
  (void)in_sizes; (void)n_in; (void)out_size; (void)ws_size;
  const float* preds   = (const float*)d_in[0];
  const float* targets = (const float*)d_in[1];
  const float* flow    = (const float*)d_in[2];
  const int*   labels  = (const int*)d_in[3];   // harness maps integer inputs -> int32

  float* pN = (float*)d_ws;                       // B*C*GRID_X floats
  float* pD = pN + (size_t)B_ * C_ * GRID_X;      // B*C*GRID_X floats
  float* r  = pD + (size_t)B_ * C_ * GRID_X;      // B*C floats

  dim3 grid(GRID_X, B_);
  tcloss_main<<<grid, 256, 0, stream>>>(preds, targets, flow, labels, pN, pD);
  tcloss_reduce<<<B_ * C_, 256, 0, stream>>>(pN, pD, r);
  tcloss_finalize<<<1, 32, 0, stream>>>(r, (float*)d_out);
}